// _SAGE_90907277787210
// MI455X (gfx1250) — compile-verified
//
#include <hip/hip_runtime.h>

// GraphSAGE 2-layer forward for MI455X (gfx1250, wave32).
//  - Aggregation: CSR-by-dst built once per launch (count -> scan -> fill),
//    then one wave32 per node gathers neighbor rows (coalesced 512B loads,
//    NO feature atomics) and writes the sum with a plain float4 store.
//  - GEMM: V_WMMA_F32_16X16X4_F32; wave owns 16 rows; K=128 A-fragments in
//    VGPRs; weight tiles (128x16 f32) DMA'd into LDS by the Tensor Data Mover
//    (tensor_load_to_lds + s_wait_tensorcnt), double-buffered ping-pong.
//  - 100000 rows = 6250 waves = 3125 blocks x 2 waves exactly => no tail;
//    layer 2 runs in-place on d_out (each wave only touches its own 16 rows).
// Workspace (~59 MB): summed | row_start | counts | excl | cursor | bsum | csr_src.

typedef __attribute__((ext_vector_type(2))) float v2f;
typedef __attribute__((ext_vector_type(8))) float v8f;
typedef __attribute__((ext_vector_type(4))) unsigned int u32x4;
typedef __attribute__((ext_vector_type(8))) int i32x8;
typedef __attribute__((ext_vector_type(4))) int i32x4;

#define NN 100000
#define NE 1600000
#define DIM 128
#define NBLK 391  // ceil(NN/256)

// ---------------- CSR build ----------------

__global__ void _zero_u32_kernel(unsigned* __restrict__ p, int n) {
  int i = blockIdx.x * blockDim.x + threadIdx.x;
  if (i < n) p[i] = 0u;
}

__global__ void _count_kernel(const long long* __restrict__ dst,
                              unsigned* __restrict__ counts) {
  int e = blockIdx.x * blockDim.x + threadIdx.x;
  if (e >= NE) return;
  atomicAdd(&counts[(unsigned)dst[e]], 1u);
}

// Per-block exclusive scan of counts; block totals to bsum.
__global__ void _scan1_kernel(const unsigned* __restrict__ counts,
                              unsigned* __restrict__ excl,
                              unsigned* __restrict__ bsum) {
  __shared__ unsigned tmp[256];
  const int t = threadIdx.x;
  const int gid = blockIdx.x * 256 + t;
  unsigned v = (gid < NN) ? counts[gid] : 0u;
  tmp[t] = v;
  __syncthreads();
  for (int off = 1; off < 256; off <<= 1) {
    unsigned add = (t >= off) ? tmp[t - off] : 0u;
    __syncthreads();
    tmp[t] += add;
    __syncthreads();
  }
  if (gid < NN) excl[gid] = tmp[t] - v;  // exclusive = inclusive - self
  if (t == 255) bsum[blockIdx.x] = tmp[255];
}

// Single block: exclusive scan of the NBLK block totals (NBLK <= 512).
__global__ void _scan2_kernel(unsigned* __restrict__ bsum) {
  __shared__ unsigned tmp[512];
  const int t = threadIdx.x;
  unsigned v = (t < NBLK) ? bsum[t] : 0u;
  tmp[t] = v;
  __syncthreads();
  for (int off = 1; off < 512; off <<= 1) {
    unsigned add = (t >= off) ? tmp[t - off] : 0u;
    __syncthreads();
    tmp[t] += add;
    __syncthreads();
  }
  if (t < NBLK) bsum[t] = tmp[t] - v;
}

__global__ void _scan3_kernel(const unsigned* __restrict__ excl,
                              const unsigned* __restrict__ bsum,
                              unsigned* __restrict__ row_start) {
  const int gid = blockIdx.x * 256 + threadIdx.x;
  if (gid < NN) row_start[gid] = excl[gid] + bsum[blockIdx.x];
  if (gid == 0) row_start[NN] = NE;
}

__global__ void _fill_kernel(const long long* __restrict__ src,
                             const long long* __restrict__ dst,
                             const unsigned* __restrict__ row_start,
                             unsigned* __restrict__ cursor,
                             unsigned* __restrict__ csr_src) {
  int e = blockIdx.x * blockDim.x + threadIdx.x;
  if (e >= NE) return;
  unsigned d = (unsigned)dst[e];
  unsigned pos = row_start[d] + atomicAdd(&cursor[d], 1u);
  csr_src[pos] = (unsigned)src[e];
}

// ---------------- Aggregation: one wave per node, no atomics ----------------

__global__ void _sage_gather_kernel(const float* __restrict__ h,
                                    const unsigned* __restrict__ row_start,
                                    const unsigned* __restrict__ csr_src,
                                    float* __restrict__ summed) {
  const unsigned gid = blockIdx.x * blockDim.x + threadIdx.x;
  const unsigned node = gid >> 5;
  const unsigned q = gid & 31;  // lane: features 4q..4q+3
  if (node >= NN) return;
  const unsigned beg = row_start[node];
  const unsigned end = row_start[node + 1];
  float4 acc = {0.0f, 0.0f, 0.0f, 0.0f};
  for (unsigned i = beg; i < end; ++i) {
    const unsigned s = csr_src[i];  // uniform across wave -> broadcast load
    const float4 v = *(const float4*)(h + (long long)s * DIM + q * 4);
    acc.x += v.x; acc.y += v.y; acc.z += v.z; acc.w += v.w;
  }
  *(float4*)(summed + (long long)node * DIM + q * 4) = acc;  // plain store
}

// ---------------- TDM weight-tile DMA ----------------

// DMA one 128x16 f32 tile (row stride 128 floats) from global into LDS.
// D# per CDNA5 ISA ch.8: group0 = {count=1, lds_addr, global_addr, type=2},
// group1 = {data_size=4B, tensor_dim0=16, tensor_dim1=128, tile_dim0=16,
// tile_dim1=128, tensor_dim0_stride=128}. Groups 2/3 zero (<=2D tensor).
__device__ __forceinline__ void tdm_load_tile_128x16(unsigned lds_byte_off,
                                                     const float* gptr) {
  unsigned long long ga = (unsigned long long)gptr;
  u32x4 g0;
  g0[0] = 1u;                                   // count=1 (valid user descriptor)
  g0[1] = lds_byte_off;                         // lds_addr [63:32]
  g0[2] = (unsigned)ga;                         // global_addr lo [95:64]
  g0[3] = (unsigned)((ga >> 32) & 0x01FFFFFFu)  // global_addr hi [120:96]
          | 0x80000000u;                        // type=2 [127:126]
  i32x8 g1;
  g1[0] = 0x00020000;        // workgroup_mask=0, data_size=2 (4 bytes)
  g1[1] = (int)(16u << 16);  // tensor_dim0 = 16    [79:48] lo
  g1[2] = (int)(128u << 16); // tensor_dim1 = 128   [111:80] lo
  g1[3] = (int)(16u << 16);  // tile_dim0 = 16      [127:112]
  g1[4] = 128;               // tile_dim1 = 128, tile_dim2 = 0
  g1[5] = 128;               // tensor_dim0_stride = 128 (lo 32 of 48)
  g1[6] = 0;
  g1[7] = 0;
  i32x4 z4 = {0, 0, 0, 0};
  i32x8 z8 = {0, 0, 0, 0, 0, 0, 0, 0};
  __builtin_amdgcn_tensor_load_to_lds(g0, g1, z4, z4, z8, 0);
}

// ---------------- WMMA GEMM ----------------

// out[i,:] = relu( (summed[i,:]/max(deg_i,1)) @ Wl + h[i,:] @ Wr + b ),
// deg_i = row_start[i+1]-row_start[i].
// Block: 64 threads = 2 waves, each wave computes 16 rows x 128 cols.
__global__ __launch_bounds__(64)
void _sage_gemm_kernel(const float* __restrict__ h,
                       const float* __restrict__ summed,
                       const unsigned* __restrict__ row_start,
                       const float* __restrict__ Wl,
                       const float* __restrict__ Wr,
                       const float* __restrict__ bias,
                       float* __restrict__ out) {
  // Ping-pong weight tiles in TDM-native layout: [buf][mat][k][n], 32 KB.
  __shared__ float ldsW[2][2][128][16];

  const int tid  = threadIdx.x;
  const int lane = tid & 31;
  const int wave = tid >> 5;
  const int m    = lane & 15;   // row-in-tile (A) / col-in-tile (B,D)
  const int kh   = lane >> 4;   // K-half select for A and B fragments

  const long long rowbase = ((long long)blockIdx.x * 2 + wave) * 16;
  const long long row     = rowbase + m;

  const unsigned rb = row_start[row];
  const unsigned re = row_start[row + 1];
  const float rdeg = 1.0f / fmaxf((float)(re - rb), 1.0f);

  // A fragments for full K=128 in VGPRs (reused over all N-tiles).
  // V_WMMA_F32_16X16X4_F32 A layout: lane m (+16*kh) holds A[m][4t+2*kh+{0,1}].
  v2f am[32], ah[32];
  {
    const float* hp = h      + row * DIM + 2 * kh;
    const float* sp = summed + row * DIM + 2 * kh;
#pragma unroll
    for (int t = 0; t < 32; ++t) {
      v2f a = *(const v2f*)(hp + 4 * t);
      v2f s = *(const v2f*)(sp + 4 * t);
      ah[t] = a;
      am[t] = s * rdeg;
    }
  }

  // Hoist bias values out of the nt loop.
  float bvals[8];
#pragma unroll
  for (int nt = 0; nt < 8; ++nt) bvals[nt] = bias[nt * 16 + m];

  // LDS byte offsets (generic ptr low 32 bits = LDS offset).
  const unsigned lds_base = (unsigned)(unsigned long long)&ldsW[0][0][0][0];

  // Prologue: wave 0 DMAs tile nt=0 (Wl and Wr) into buffer 0.
  if (wave == 0) {
    tdm_load_tile_128x16(lds_base + 0 * 8192, Wl);
    tdm_load_tile_128x16(lds_base + 1 * 8192, Wr);
  }

#pragma unroll 1
  for (int nt = 0; nt < 8; ++nt) {
    const int buf = nt & 1;
    __syncthreads();  // readers of buf[(nt+1)&1] (iter nt-1) are done
    if (wave == 0) {
      if (nt < 7) {
        const unsigned nb = (nt + 1) & 1;
        tdm_load_tile_128x16(lds_base + (nb * 2 + 0) * 8192, Wl + (nt + 1) * 16);
        tdm_load_tile_128x16(lds_base + (nb * 2 + 1) * 8192, Wr + (nt + 1) * 16);
        __builtin_amdgcn_s_wait_tensorcnt(2);  // in-order => tile nt complete
      } else {
        __builtin_amdgcn_s_wait_tensorcnt(0);
      }
    }
    __syncthreads();  // publish buf[nt&1] to both waves

    const float (*tileL)[16] = ldsW[buf][0];
    const float (*tileR)[16] = ldsW[buf][1];

    const float bb = bvals[nt];
    v8f acc = {bb, bb, bb, bb, bb, bb, bb, bb};

    // D(16x16) += A_mean @ Wl_tile + A_h @ Wr_tile, 32 K-steps of 4.
    // B fragment: lane (m,kh) needs W[k0+2kh+{0,1}][m] -> ds_load_2addr_b32,
    // conflict-free (bank (16k+m)%64 distinct per lane).
#pragma unroll
    for (int t = 0; t < 32; ++t) {
      const int k0 = 4 * t + 2 * kh;
      v2f bl = v2f{tileL[k0][m], tileL[k0 + 1][m]};
      v2f br = v2f{tileR[k0][m], tileR[k0 + 1][m]};
      acc = __builtin_amdgcn_wmma_f32_16x16x4_f32(
          false, am[t], false, bl, (short)0, acc, false, false);
      acc = __builtin_amdgcn_wmma_f32_16x16x4_f32(
          false, ah[t], false, br, (short)0, acc, false, false);
    }

    // C/D layout: VGPR r -> row r (lanes 0-15) / r+8 (lanes 16-31); col = n0+m.
#pragma unroll
    for (int r = 0; r < 8; ++r) {
      const long long orow = rowbase + r + 8 * kh;
      out[orow * DIM + nt * 16 + m] = fmaxf(acc[r], 0.0f);
    }
  }
}

// ---------------- Host launch ----------------

extern "C" void kernel_launch(void* const* d_in, const int* in_sizes, int n_in,
                              void* d_out, int out_size, void* d_ws, size_t ws_size,
                              hipStream_t stream) {
  const float*     x   = (const float*)d_in[0];
  const long long* ei  = (const long long*)d_in[1];  // int64 [2, NE]
  const float*     Wl0 = (const float*)d_in[2];
  const float*     Wr0 = (const float*)d_in[3];
  const float*     b0  = (const float*)d_in[4];
  const float*     Wl1 = (const float*)d_in[5];
  const float*     Wr1 = (const float*)d_in[6];
  const float*     b1  = (const float*)d_in[7];
  float* out = (float*)d_out;

  // Workspace layout (16B-aligned slices).
  char* ws = (char*)d_ws;
  float*    summed    = (float*)ws;                    ws += (size_t)NN * DIM * 4;  // 51.2 MB
  unsigned* row_start = (unsigned*)ws;                 ws += (size_t)(NN + 4) * 4;
  unsigned* counts    = (unsigned*)ws;                 ws += (size_t)NN * 4;
  unsigned* excl      = (unsigned*)ws;                 ws += (size_t)NN * 4;
  unsigned* cursor    = (unsigned*)ws;                 ws += (size_t)NN * 4;
  unsigned* bsum      = (unsigned*)ws;                 ws += (size_t)512 * 4;
  unsigned* csr_src   = (unsigned*)ws;                 // NE * 4 = 6.4 MB

  const long long* src = ei;
  const long long* dst = ei + NE;

  const int edge_grid = NE / 256;       // 6250 exact
  const int node_grid = (NN + 255) / 256;
  const int gath_grid = NN * 32 / 256;  // 12500 exact (wave per node)
  const int gemm_grid = NN / 32;        // 3125 blocks * 2 waves * 16 rows

  // ---- CSR build (by dst), once, reused by both layers ----
  _zero_u32_kernel<<<node_grid, 256, 0, stream>>>(counts, NN);
  _zero_u32_kernel<<<node_grid, 256, 0, stream>>>(cursor, NN);
  _count_kernel<<<edge_grid, 256, 0, stream>>>(dst, counts);
  _scan1_kernel<<<NBLK, 256, 0, stream>>>(counts, excl, bsum);
  _scan2_kernel<<<1, 512, 0, stream>>>(bsum);
  _scan3_kernel<<<NBLK, 256, 0, stream>>>(excl, bsum, row_start);
  _fill_kernel<<<edge_grid, 256, 0, stream>>>(src, dst, row_start, cursor, csr_src);

  // ---- Layer 0: out = relu(sage(x)) ----
  _sage_gather_kernel<<<gath_grid, 256, 0, stream>>>(x, row_start, csr_src, summed);
  _sage_gemm_kernel<<<gemm_grid, 64, 0, stream>>>(x, summed, row_start, Wl0, Wr0, b0, out);

  // ---- Layer 1: out = relu(sage(out)) in-place ----
  _sage_gather_kernel<<<gath_grid, 256, 0, stream>>>(out, row_start, csr_src, summed);
  _sage_gemm_kernel<<<gemm_grid, 64, 0, stream>>>(out, summed, row_start, Wl1, Wr1, b1, out);
}